// ConnectLayer_33466385171003
// MI455X (gfx1250) — compile-verified
//
#include <hip/hip_runtime.h>
#include <hip/hip_bf16.h>

// Problem dims (fixed by the reference)
#define S_DIM 128
#define B_DIM 64
#define W_DIM 32
#define D2    512
#define DH    512
#define K2    (2 * D2)   // 1024

typedef __attribute__((ext_vector_type(2))) float v2f;
typedef __attribute__((ext_vector_type(8))) float v8f;

// ---------------------------------------------------------------------------
// Kernel 1: wiki_cv gather + build X = [h_n | wiki_cv]  (B x 1024) in ws
// grid: B blocks x 128 threads; each thread moves one float4 per half-row
// ---------------------------------------------------------------------------
__global__ void build_x_kernel(const float* __restrict__ h_n,
                               const float* __restrict__ h_n1,
                               const int* __restrict__ valid_sen,
                               const int* __restrict__ atten_label,
                               float* __restrict__ X,
                               float* __restrict__ out_wiki_cv) {
    const int b = blockIdx.x;
    const int t = threadIdx.x;               // 0..127, 128 float4 = 512 floats
    const int vb = valid_sen[b];
    const int w  = atten_label[vb];

    const float4* hn = (const float4*)(h_n  + (size_t)b * D2);
    const float4* cv = (const float4*)(h_n1 + ((size_t)w * B_DIM + vb) * D2);
    float4* Xr = (float4*)(X + (size_t)b * K2);
    float4* wc = (float4*)(out_wiki_cv + (size_t)b * D2);

    Xr[t] = hn[t];
    float4 v = cv[t];
    Xr[128 + t] = v;
    wc[t] = v;
}

// ---------------------------------------------------------------------------
// Kernel 2: beta[b,w] = dot(h_n1[w, vs[b], :], h_n[b, :])
// grid: B*W blocks of 32 threads (1 wave each)
// ---------------------------------------------------------------------------
__global__ void beta_kernel(const float* __restrict__ h_n1,
                            const float* __restrict__ h_n,
                            const int* __restrict__ valid_sen,
                            float* __restrict__ beta) {
    const int w = blockIdx.x & (W_DIM - 1);
    const int b = blockIdx.x >> 5;
    const int lane = threadIdx.x;            // 0..31
    const int vb = valid_sen[b];

    const float4* a = (const float4*)(h_n1 + ((size_t)w * B_DIM + vb) * D2);
    const float4* x = (const float4*)(h_n + (size_t)b * D2);

    float acc = 0.f;
#pragma unroll
    for (int i = lane; i < D2 / 4; i += 32) {
        float4 av = a[i], xv = x[i];
        acc += av.x * xv.x + av.y * xv.y + av.z * xv.z + av.w * xv.w;
    }
#pragma unroll
    for (int m = 16; m; m >>= 1) acc += __shfl_xor(acc, m, 32);
    if (lane == 0) beta[b * W_DIM + w] = acc;
}

// ---------------------------------------------------------------------------
// Kernel 3: atten_loss. 1 block x 64 threads (thread b = one batch row)
// ---------------------------------------------------------------------------
__global__ void loss_kernel(const float* __restrict__ beta,
                            const int* __restrict__ valid_sen,
                            const int* __restrict__ atten_label,
                            float* __restrict__ out_loss) {
    __shared__ float s_nll[B_DIM];
    __shared__ int   s_cnt[B_DIM];
    const int b = threadIdx.x;
    const int vb = valid_sen[b];
    const int lbl = atten_label[vb];
    const float* row = beta + b * W_DIM;

    float mx = row[0];
#pragma unroll
    for (int w = 1; w < W_DIM; ++w) mx = fmaxf(mx, row[w]);
    float s = 0.f;
#pragma unroll
    for (int w = 0; w < W_DIM; ++w) s += expf(row[w] - mx);
    const float nll = -(row[lbl] - mx - logf(s));
    const bool valid = (lbl != 0);
    s_nll[b] = valid ? nll : 0.f;
    s_cnt[b] = valid ? 1 : 0;
    __syncthreads();
#pragma unroll
    for (int off = B_DIM / 2; off; off >>= 1) {
        if (b < off) { s_nll[b] += s_nll[b + off]; s_cnt[b] += s_cnt[b + off]; }
        __syncthreads();
    }
    if (b == 0) {
        const float cnt = (float)s_cnt[0];
        out_loss[0] = (cnt > 0.f) ? (s_nll[0] / fmaxf(cnt, 1.f)) : 0.f;
    }
}

// ---------------------------------------------------------------------------
// Kernel 4: the dominant 16MB gather: selected_wiki_h + selected_wiki_sen
// grid: B*S blocks x 128 threads; one b128 load + one b128 store per thread
// ---------------------------------------------------------------------------
__global__ void gather_kernel(const float* __restrict__ h1,
                              const int* __restrict__ wiki_sen,
                              const int* __restrict__ valid_sen,
                              const int* __restrict__ atten_label,
                              float* __restrict__ out_sen,
                              float* __restrict__ out_swh) {
    const int bs = blockIdx.x;
    const int b = bs >> 7;                   // /S_DIM
    const int s = bs & (S_DIM - 1);
    const int t = threadIdx.x;               // 0..127
    const int vb = valid_sen[b];
    const int w  = atten_label[vb];

    const float4* src = (const float4*)(h1 + (((size_t)s * B_DIM + vb) * W_DIM + w) * D2);
    float4* dst = (float4*)(out_swh + ((size_t)b * S_DIM + s) * D2);
    dst[t] = src[t];

    if (t == 0) {
        out_sen[b * S_DIM + s] =
            (float)wiki_sen[((size_t)vb * W_DIM + w) * S_DIM + s];
    }
}

// ---------------------------------------------------------------------------
// Kernel 5: init_h = X @ W_init.T + b_init via V_WMMA_F32_16X16X4_F32
// 128 waves total: 32 blocks x 128 threads (4 waves/block).
// wave -> (mtile = wave & 3, ntile = wave >> 2); K loop 1024 in steps of 4.
// A-frag layout (32-bit 16x4): lane L holds M = L%16, VGPR r holds
// K = r + 2*(L/16)  -> one aligned float2 load per step. Same for B (N-major).
// D layout: VGPR r -> row r + 8*(L/16), col L%16.
// ---------------------------------------------------------------------------
__global__ void init_h_wmma_kernel(const float* __restrict__ X,
                                   const float* __restrict__ Wm,
                                   const float* __restrict__ b_init,
                                   float* __restrict__ out) {
    const int lane  = threadIdx.x & 31;
    const int wave  = blockIdx.x * 4 + (threadIdx.x >> 5);
    const int mtile = wave & 3;              // 4 tiles cover M=64
    const int ntile = wave >> 2;             // 32 tiles cover N=512
    const int m = mtile * 16 + (lane & 15);
    const int n = ntile * 16 + (lane & 15);
    const int koff = (lane >> 4) << 1;       // 0 or 2

    const v2f* Xr = (const v2f*)(X + (size_t)m * K2);
    const v2f* Wr = (const v2f*)(Wm + (size_t)n * K2);

    v8f c = {};
#pragma unroll 8
    for (int k = 0; k < K2; k += 4) {
        v2f a = Xr[(k + koff) >> 1];
        v2f bb = Wr[(k + koff) >> 1];
        c = __builtin_amdgcn_wmma_f32_16x16x4_f32(
                /*neg_a=*/false, a, /*neg_b=*/false, bb,
                /*c_mod=*/(short)0, c, /*reuse_a=*/false, /*reuse_b=*/false);
    }

    const float bias = b_init[n];
    const int rbase = mtile * 16 + ((lane >> 4) << 3);
#pragma unroll
    for (int r = 0; r < 8; ++r) {
        out[(size_t)(rbase + r) * DH + n] = c[r] + bias;
    }
}

// ---------------------------------------------------------------------------
extern "C" void kernel_launch(void* const* d_in, const int* in_sizes, int n_in,
                              void* d_out, int out_size, void* d_ws, size_t ws_size,
                              hipStream_t stream) {
    // Inputs in setup_inputs() order:
    // 0:i 1:valid_sen 2:reverse_valid_sen 3:wiki_sen 4:wiki_num
    // 5:h1 6:h_n1 7:atten_label 8:h 9:h_n 10:W_init 11:b_init
    const int*   valid_sen   = (const int*)d_in[1];
    const int*   wiki_sen    = (const int*)d_in[3];
    const float* h1          = (const float*)d_in[5];
    const float* h_n1        = (const float*)d_in[6];
    const int*   atten_label = (const int*)d_in[7];
    const float* h_n         = (const float*)d_in[9];
    const float* W_init      = (const float*)d_in[10];
    const float* b_init      = (const float*)d_in[11];

    float* out = (float*)d_out;
    // Output regions (floats), concatenated in reference return order:
    float* out_sen     = out;                                   // B*S       = 8192
    float* out_swh     = out + (size_t)B_DIM * S_DIM;           // B*S*D2    = 4194304
    float* out_init_h  = out_swh + (size_t)B_DIM * S_DIM * D2;  // B*DH      = 32768
    float* out_wiki_cv = out_init_h + (size_t)B_DIM * DH;       // B*D2      = 32768
    float* out_loss    = out_wiki_cv + (size_t)B_DIM * D2;      // 1

    float* ws   = (float*)d_ws;
    float* beta = ws;                         // B*W = 2048 floats
    float* X    = ws + 2048;                  // B*K2 = 65536 floats (256KB)

    // 1) wiki_cv + X = [h_n | wiki_cv]
    build_x_kernel<<<B_DIM, 128, 0, stream>>>(h_n, h_n1, valid_sen, atten_label,
                                              X, out_wiki_cv);
    // 2) beta
    beta_kernel<<<B_DIM * W_DIM, 32, 0, stream>>>(h_n1, h_n, valid_sen, beta);
    // 3) loss
    loss_kernel<<<1, B_DIM, 0, stream>>>(beta, valid_sen, atten_label, out_loss);
    // 4) big gather (dominant memory traffic)
    gather_kernel<<<B_DIM * S_DIM, 128, 0, stream>>>(h1, wiki_sen, valid_sen,
                                                     atten_label, out_sen, out_swh);
    // 5) init_h GEMM on the WMMA pipe (fp32-exact: 16x16x4 f32)
    init_h_wmma_kernel<<<32, 128, 0, stream>>>(X, W_init, b_init, out_init_h);
}